// TemporalGCNClassifierO_48996986912815
// MI455X (gfx1250) — compile-verified
//
#include <hip/hip_runtime.h>
#include <hip/hip_bf16.h>
#include <math.h>

#define F 128

typedef __attribute__((ext_vector_type(2))) float v2f;
typedef __attribute__((ext_vector_type(8))) float v8f;

__device__ __forceinline__ float sigmoid_f(float x) {
  return 1.0f / (1.0f + __expf(-x));
}

// ---------------------------------------------------------------------------
// Kernel 1: EvolveGCN-O single LSTM step on the FxF weight (h0 = c0 = 0).
// gates = W0 @ w_ih^T + b_ih + b_hh ; W = sigmoid(o) * tanh(sigmoid(i)*tanh(g))
// 16384 threads, ~12.6 MFLOP total -> VALU is fine, not worth the matrix path.
// ---------------------------------------------------------------------------
__global__ __launch_bounds__(256) void k_evolve(const float* __restrict__ W0,
                                                const float* __restrict__ w_ih,
                                                const float* __restrict__ b_ih,
                                                const float* __restrict__ b_hh,
                                                float* __restrict__ W) {
  int t = blockIdx.x * blockDim.x + threadIdx.x;  // 0 .. F*F-1
  int r = t >> 7;
  int c = t & (F - 1);
  float ai = b_ih[c]         + b_hh[c];           // input gate (j = c)
  float ag = b_ih[2 * F + c] + b_hh[2 * F + c];   // cell gate  (j = 2F + c)
  float ao = b_ih[3 * F + c] + b_hh[3 * F + c];   // out gate   (j = 3F + c)
  const float* wi = w_ih + (size_t)c * F;
  const float* wg = w_ih + (size_t)(2 * F + c) * F;
  const float* wo = w_ih + (size_t)(3 * F + c) * F;
  const float* w0 = W0 + (size_t)r * F;
#pragma unroll 4
  for (int k = 0; k < F; ++k) {
    float v = w0[k];
    ai = fmaf(v, wi[k], ai);
    ag = fmaf(v, wg[k], ag);
    ao = fmaf(v, wo[k], ao);
  }
  float cs = sigmoid_f(ai) * tanhf(ag);
  W[t] = sigmoid_f(ao) * tanhf(cs);
}

// ---------------------------------------------------------------------------
// Kernel 2/3/4: degree with self-loops -> dis = rsqrt(deg)
// ---------------------------------------------------------------------------
__global__ __launch_bounds__(256) void k_deg_init(float* __restrict__ deg, int n) {
  int i = blockIdx.x * blockDim.x + threadIdx.x;
  if (i < n) deg[i] = 1.0f;  // self loop contributes 1 to every node
}

__global__ __launch_bounds__(256) void k_deg_acc(const int* __restrict__ col,
                                                 float* __restrict__ deg,
                                                 int e, int n) {
  int i = blockIdx.x * blockDim.x + threadIdx.x;
  if (i < e) {
    int t = col[i];
    if ((unsigned)t < (unsigned)n) atomicAdd(&deg[t], 1.0f);
  }
}

__global__ __launch_bounds__(256) void k_rsqrt(float* __restrict__ deg, int n) {
  int i = blockIdx.x * blockDim.x + threadIdx.x;
  if (i < n) deg[i] = rsqrtf(deg[i]);  // deg >= 1 always (self loops)
}

// ---------------------------------------------------------------------------
// Kernel 5: xw = x @ W via V_WMMA_F32_16X16X4_F32 (wave32 matrix path).
// One 256-thread block = 8 waves handles a 16-row strip of x; wave w owns
// column tile [16w, 16w+16). K loop: 128/4 = 32 WMMA issues per tile.
// fp32 A 16x4 layout: lanes 0-15 -> (M=lane, K=0/1), lanes 16-31 -> (M, K=2/3).
// fp32 B 4x16 layout: lanes 0-15 -> (N=lane, K=0/1), lanes 16-31 -> (N, K=2/3).
// C/D 16x16: VGPR r -> row m0+r (lanes 0-15) / m0+8+r (lanes 16-31), N=lane&15.
// ---------------------------------------------------------------------------
__global__ __launch_bounds__(256) void k_gemm_xw(const float* __restrict__ x,
                                                 const float* __restrict__ W,
                                                 float* __restrict__ xw, int n) {
  const int lane = threadIdx.x & 31;
  const int wave = threadIdx.x >> 5;   // 0..7  -> column tile
  const int m0   = blockIdx.x * 16;    // row strip
  const int n0   = wave * 16;
  const int half = lane >> 4;          // 0: lanes 0-15, 1: lanes 16-31
  const int l15  = lane & 15;

  int arow = m0 + l15;
  if (arow >= n) arow = n - 1;         // clamp (keeps EXEC all-1s for WMMA)
  const int bcol = n0 + l15;

  v8f acc = {};
#pragma unroll
  for (int k = 0; k < F; k += 4) {
    const int ka = k + 2 * half;       // this half-wave's K pair
    v2f a, b;
    a.x = x[(size_t)arow * F + ka];
    a.y = x[(size_t)arow * F + ka + 1];
    b.x = W[(size_t)ka * F + bcol];
    b.y = W[(size_t)(ka + 1) * F + bcol];
    acc = __builtin_amdgcn_wmma_f32_16x16x4_f32(
        /*neg_a=*/false, a, /*neg_b=*/false, b,
        /*c_mod=*/(short)0, acc, /*reuse_a=*/false, /*reuse_b=*/false);
  }

  const int col = n0 + l15;
#pragma unroll
  for (int r = 0; r < 8; ++r) {
    int row = m0 + r + 8 * half;
    if (row < n) xw[(size_t)row * F + col] = acc[r];
  }
}

// ---------------------------------------------------------------------------
// Kernel 6: self-loop init  h = dis[n]^2 * xw   (float4 per thread)
// ---------------------------------------------------------------------------
__global__ __launch_bounds__(256) void k_selfloop(const float4* __restrict__ xw4,
                                                  const float* __restrict__ dis,
                                                  float4* __restrict__ h4, int n) {
  int t = blockIdx.x * blockDim.x + threadIdx.x;  // 0 .. n*32-1
  if (t < n * (F / 4)) {
    int node = t >> 5;
    float s = dis[node];
    s = s * s;
    float4 v = xw4[t];
    v.x *= s; v.y *= s; v.z *= s; v.w *= s;
    h4[t] = v;
  }
}

// ---------------------------------------------------------------------------
// Kernel 7: edge scatter  h[col] += dis[row]*dis[col] * xw[row]
// One wave32 per edge; each lane moves 4 floats (float4 gather + 4 atomics).
// xw and h are both L2-resident (25.6 MB each vs 192 MB L2).
// ---------------------------------------------------------------------------
__global__ __launch_bounds__(256) void k_scatter(const int* __restrict__ row,
                                                 const int* __restrict__ col,
                                                 const float* __restrict__ dis,
                                                 const float4* __restrict__ xw4,
                                                 float* __restrict__ h,
                                                 int e, int n) {
  int eidx = (blockIdx.x * blockDim.x + threadIdx.x) >> 5;
  int lane = threadIdx.x & 31;
  if (eidx >= e) return;
  int s = row[eidx];
  int t = col[eidx];
  if ((unsigned)s >= (unsigned)n || (unsigned)t >= (unsigned)n) return;
  float nrm = dis[s] * dis[t];
  float4 v = xw4[(size_t)s * (F / 4) + lane];
  float* dst = h + (size_t)t * F + lane * 4;
  atomicAdd(dst + 0, nrm * v.x);
  atomicAdd(dst + 1, nrm * v.y);
  atomicAdd(dst + 2, nrm * v.z);
  atomicAdd(dst + 3, nrm * v.w);
}

// ---------------------------------------------------------------------------
// Kernel 8: classifier  out = relu(h + gcn_bias) @ Wc^T + bc   (C = 2)
// One wave per node; lane handles 4 features, then wave32 shuffle reduction.
// ---------------------------------------------------------------------------
__global__ __launch_bounds__(256) void k_classify(const float4* __restrict__ h4,
                                                  const float4* __restrict__ bias4,
                                                  const float4* __restrict__ Wc4,
                                                  const float* __restrict__ bc,
                                                  float* __restrict__ out, int n) {
  int node = (blockIdx.x * blockDim.x + threadIdx.x) >> 5;
  int lane = threadIdx.x & 31;
  if (node >= n) return;
  float4 v = h4[(size_t)node * (F / 4) + lane];
  float4 b = bias4[lane];
  float4 w0 = Wc4[lane];              // Wc row 0
  float4 w1 = Wc4[(F / 4) + lane];    // Wc row 1
  float r0, r1, a0 = 0.0f, a1 = 0.0f;
  r0 = fmaxf(v.x + b.x, 0.0f); a0 = fmaf(r0, w0.x, a0); a1 = fmaf(r0, w1.x, a1);
  r0 = fmaxf(v.y + b.y, 0.0f); a0 = fmaf(r0, w0.y, a0); a1 = fmaf(r0, w1.y, a1);
  r0 = fmaxf(v.z + b.z, 0.0f); a0 = fmaf(r0, w0.z, a0); a1 = fmaf(r0, w1.z, a1);
  r1 = fmaxf(v.w + b.w, 0.0f); a0 = fmaf(r1, w0.w, a0); a1 = fmaf(r1, w1.w, a1);
#pragma unroll
  for (int off = 16; off > 0; off >>= 1) {
    a0 += __shfl_down(a0, off, 32);
    a1 += __shfl_down(a1, off, 32);
  }
  if (lane == 0) {
    out[(size_t)node * 2 + 0] = a0 + bc[0];
    out[(size_t)node * 2 + 1] = a1 + bc[1];
  }
}

// ---------------------------------------------------------------------------
// Launch
// ---------------------------------------------------------------------------
extern "C" void kernel_launch(void* const* d_in, const int* in_sizes, int n_in,
                              void* d_out, int out_size, void* d_ws, size_t ws_size,
                              hipStream_t stream) {
  const float* x        = (const float*)d_in[0];  // [N, F]
  const float* W0       = (const float*)d_in[1];  // [F, F]
  const float* w_ih     = (const float*)d_in[2];  // [4F, F]
  // d_in[3] = w_hh: unused (h0 = 0)
  const float* b_ih     = (const float*)d_in[4];  // [4F]
  const float* b_hh     = (const float*)d_in[5];  // [4F]
  const float* gcn_bias = (const float*)d_in[6];  // [F]
  const float* Wc       = (const float*)d_in[7];  // [C, F]
  const float* bc       = (const float*)d_in[8];  // [C]
  const int*   ei       = (const int*)d_in[9];    // [2, E]
  float* out = (float*)d_out;

  const int n = in_sizes[0] / F;      // 50000
  const int e = in_sizes[9] / 2;      // 600000
  const int* erow = ei;               // edge_index[0] (source)
  const int* ecol = ei + e;           // edge_index[1] (target)

  // Workspace layout (floats):
  //   W   : F*F           = 16384
  //   dis : n             = 50000
  //   xw  : n*F           = 6.4M   (offset 66384 floats -> 16B aligned)
  //   h   : n*F           = 6.4M   (16B aligned)
  float* W   = (float*)d_ws;
  float* dis = W + F * F;
  float* xw  = dis + n;
  float* h   = xw + (size_t)n * F;

  const int BT = 256;

  // 1) evolve GCN weight
  k_evolve<<<(F * F + BT - 1) / BT, BT, 0, stream>>>(W0, w_ih, b_ih, b_hh, W);

  // 2-4) degrees -> dis
  k_deg_init<<<(n + BT - 1) / BT, BT, 0, stream>>>(dis, n);
  k_deg_acc<<<(e + BT - 1) / BT, BT, 0, stream>>>(ecol, dis, e, n);
  k_rsqrt<<<(n + BT - 1) / BT, BT, 0, stream>>>(dis, n);

  // 5) xw = x @ W (WMMA fp32 path)
  k_gemm_xw<<<(n + 15) / 16, BT, 0, stream>>>(x, W, xw, n);

  // 6) h = dis^2 * xw  (self loops)
  k_selfloop<<<((size_t)n * (F / 4) + BT - 1) / BT, BT, 0, stream>>>(
      (const float4*)xw, dis, (float4*)h, n);

  // 7) edge scatter (one wave per edge)
  {
    long long waves = e;
    long long blocks = (waves * 32 + BT - 1) / BT;
    k_scatter<<<(int)blocks, BT, 0, stream>>>(erow, ecol, dis,
                                              (const float4*)xw, h, e, n);
  }

  // 8) classifier (one wave per node)
  {
    long long blocks = ((long long)n * 32 + BT - 1) / BT;
    k_classify<<<(int)blocks, BT, 0, stream>>>((const float4*)h,
                                               (const float4*)gcn_bias,
                                               (const float4*)Wc, bc, out, n);
  }
}